// STM_44822278701610
// MI455X (gfx1250) — compile-verified
//
#include <hip/hip_runtime.h>
#include <math.h>

typedef __attribute__((ext_vector_type(2))) float v2f;
typedef __attribute__((ext_vector_type(4))) float v4f;
typedef __attribute__((ext_vector_type(8))) float v8f;

#define B_ROWS 2048
#define D_DIM  256
#define N_COLS 16384
// GEMM block tile
#define BM 128
#define BN 256
#define BK 32
#define LDA_PAD 36   // padded LDS stride (floats) for A tile -> conflict-free strided reads

// ---------------------------------------------------------------------------
// kn[j] = sum_k kernel[k][j]^2   (column norms of (256,16384) row-major)
// ---------------------------------------------------------------------------
__global__ void __launch_bounds__(256)
som_col_norms(const float* __restrict__ K, float* __restrict__ kn) {
    int j = blockIdx.x * 256 + threadIdx.x;
    float s = 0.0f;
#pragma unroll 8
    for (int r = 0; r < D_DIM; ++r) {
        float v = K[(size_t)r * N_COLS + j];
        s += v * v;
    }
    kn[j] = s;
}

// ---------------------------------------------------------------------------
// xn[i] = sum_k x[i][k]^2   (row norms of (2048,256) row-major), one wave/row
// ---------------------------------------------------------------------------
__global__ void __launch_bounds__(256)
som_row_norms(const float* __restrict__ X, float* __restrict__ xn) {
    int wave = threadIdx.x >> 5;
    int lane = threadIdx.x & 31;
    int row  = blockIdx.x * 8 + wave;
    const float* xr = X + (size_t)row * D_DIM;
    float s = 0.0f;
#pragma unroll
    for (int c = lane; c < D_DIM; c += 32) {
        float v = xr[c];
        s += v * v;
    }
#pragma unroll
    for (int off = 16; off > 0; off >>= 1)
        s += __shfl_down(s, off, 32);
    if (lane == 0) xn[row] = s;
}

// ---------------------------------------------------------------------------
// norms2 = max(xn - 2*(x@K) + kn, 0)  ->  d_out
// 256 threads = 8 waves; block tile 128(M) x 256(N); K staged in 32-deep LDS
// chunks. Wave (waveM, waveN) in a 2x4 grid computes a 64x64 sub-tile as
// sixteen 16x16 WMMA accumulators (4 M-frags x 4 N-frags): each k-step is
// 8 fragment LDS loads feeding 16 back-to-back v_wmma_f32_16x16x4_f32.
// ---------------------------------------------------------------------------
__global__ void __launch_bounds__(256)
som_gemm_norms(const float* __restrict__ X, const float* __restrict__ K,
               const float* __restrict__ xn, const float* __restrict__ kn,
               float* __restrict__ out) {
    __shared__ float lA[BM * LDA_PAD];  // 128*36*4 = 18 KB
    __shared__ float lB[BK * BN];       // 32*256*4 = 32 KB

    const int tid   = threadIdx.x;
    const int wave  = tid >> 5;
    const int waveM = wave >> 2;        // 0..1  -> 64-row strip
    const int waveN = wave & 3;         // 0..3  -> 64-col strip
    const int lane  = tid & 31;
    const int lhalf = lane >> 4;        // 0: lanes 0-15, 1: lanes 16-31
    const int l16   = lane & 15;
    const int m_blk = blockIdx.y * BM;
    const int n_blk = blockIdx.x * BN;

    v8f acc[4][4];                      // [msub][nsub], 128 VGPRs
    v8f zero = {};
#pragma unroll
    for (int ms = 0; ms < 4; ++ms)
#pragma unroll
        for (int s = 0; s < 4; ++s) acc[ms][s] = zero;

    for (int k0 = 0; k0 < D_DIM; k0 += BK) {
        // stage A tile: 128 x 32 floats (4 x float4 per thread), padded stride
#pragma unroll
        for (int t = 0; t < 4; ++t) {
            int i = tid + t * 256;            // 0..1023
            int r = i >> 3;                   // 8 float4 per 32-float row
            int c = (i & 7) << 2;
            v4f v = *(const v4f*)(X + (size_t)(m_blk + r) * D_DIM + k0 + c);
            *(v4f*)(&lA[r * LDA_PAD + c]) = v;   // (r*144 + c*4) is 16B aligned
        }
        // stage B tile: 32 x 256 floats (8 x float4 per thread)
#pragma unroll
        for (int t = 0; t < 8; ++t) {
            int i = tid + t * 256;
            int r = i >> 6;                   // 64 float4 per 256-float row
            int c = (i & 63) << 2;
            *(v4f*)(&lB[r * BN + c]) =
                *(const v4f*)(K + (size_t)(k0 + r) * N_COLS + n_blk + c);
        }
        // prefetch next chunk's tiles while this chunk computes
        if (k0 + BK < D_DIM) {
            __builtin_prefetch(
                X + (size_t)(m_blk + (tid >> 1)) * D_DIM + (k0 + BK) + ((tid & 1) << 4), 0, 3);
            __builtin_prefetch(
                K + (size_t)(k0 + BK + (tid >> 3)) * N_COLS + n_blk + ((tid & 7) << 5), 0, 3);
        }
        __syncthreads();

#pragma unroll
        for (int kk = 0; kk < BK; kk += 4) {
            const int ka = kk + lhalf * 2;    // ISA A layout: lanes 16-31 hold K+2
            // A fragments: 16x4, lane = row M, v0/v1 = K(ka), K(ka+1)
            v2f a[4];
#pragma unroll
            for (int ms = 0; ms < 4; ++ms)
                a[ms] = *(const v2f*)(&lA[(waveM * 64 + ms * 16 + l16) * LDA_PAD + ka]);
            v2f b[4];
#pragma unroll
            for (int s = 0; s < 4; ++s) {
                const int n_loc = waveN * 64 + s * 16 + l16;
                b[s].x = lB[(ka)     * BN + n_loc];   // B 4x16: row K(ka), col = lane
                b[s].y = lB[(ka + 1) * BN + n_loc];
            }
#pragma unroll
            for (int ms = 0; ms < 4; ++ms)
#pragma unroll
                for (int s = 0; s < 4; ++s)
                    acc[ms][s] = __builtin_amdgcn_wmma_f32_16x16x4_f32(
                        false, a[ms], false, b[s], (short)0, acc[ms][s], false, false);
        }
        __syncthreads();
    }

    // Epilogue: C/D layout -> VGPR r holds row (r + lhalf*8), col = l16
    const int n_base = n_blk + waveN * 64;
#pragma unroll
    for (int ms = 0; ms < 4; ++ms) {
        const int m_base = m_blk + waveM * 64 + ms * 16 + lhalf * 8;
        float xr[8];
#pragma unroll
        for (int r = 0; r < 8; ++r) xr[r] = xn[m_base + r];
#pragma unroll
        for (int s = 0; s < 4; ++s) {
            const int n  = n_base + s * 16 + l16;
            const float knv = kn[n];
#pragma unroll
            for (int r = 0; r < 8; ++r) {
                float v = xr[r] - 2.0f * acc[ms][s][r] + knv;
                v = fmaxf(v, 0.0f);
                out[(size_t)(m_base + r) * N_COLS + n] = v;
            }
        }
    }
}

// ---------------------------------------------------------------------------
// Per sample row: argmin over 16384 units (first-index tie-break), then
// in-place multiply by Gaussian neighbourhood exp(-0.125 * grid_dist2).
// ---------------------------------------------------------------------------
__global__ void __launch_bounds__(256)
som_radials(float* __restrict__ out) {
    __shared__ float smin[256];
    __shared__ int   sidx[256];
    const int row = blockIdx.x;
    float* o = out + (size_t)row * N_COLS;
    const int tid = threadIdx.x;

    float bmin = INFINITY;
    int   bidx = 0;
    for (int j = tid; j < N_COLS; j += 256) {  // ascending j -> keeps first min
        float v = o[j];
        if (v < bmin) { bmin = v; bidx = j; }
    }
    smin[tid] = bmin;
    sidx[tid] = bidx;
    __syncthreads();
#pragma unroll
    for (int off = 128; off > 0; off >>= 1) {
        if (tid < off) {
            float v2 = smin[tid + off];
            int   i2 = sidx[tid + off];
            if (v2 < smin[tid] || (v2 == smin[tid] && i2 < sidx[tid])) {
                smin[tid] = v2;
                sidx[tid] = i2;
            }
        }
        __syncthreads();
    }
    const int wta = sidx[0];
    const float mr = (float)(wta >> 7);    // wta / 128
    const float mc = (float)(wta & 127);   // wta % 128

    for (int j = tid; j < N_COLS; j += 256) {
        float gr = (float)(j >> 7);
        float gc = (float)(j & 127);
        float dr = gr - mr, dc = gc - mc;
        o[j] *= __expf(-0.125f * (dr * dr + dc * dc));  // 0.5/sigma^2, sigma=2
    }
}

// ---------------------------------------------------------------------------
extern "C" void kernel_launch(void* const* d_in, const int* in_sizes, int n_in,
                              void* d_out, int out_size, void* d_ws, size_t ws_size,
                              hipStream_t stream) {
    const float* x   = (const float*)d_in[0];  // (2048, 256)
    const float* ker = (const float*)d_in[1];  // (256, 16384)
    float* out = (float*)d_out;                // (2048, 16384)

    float* xn = (float*)d_ws;                  // 2048 floats
    float* kn = xn + B_ROWS;                   // 16384 floats (total 72 KB of ws)

    som_row_norms<<<B_ROWS / 8, 256, 0, stream>>>(x, xn);
    som_col_norms<<<N_COLS / 256, 256, 0, stream>>>(ker, kn);

    dim3 grid(N_COLS / BN, B_ROWS / BM);       // (64, 16)
    som_gemm_norms<<<grid, 256, 0, stream>>>(x, ker, xn, kn, out);

    som_radials<<<B_ROWS, 256, 0, stream>>>(out);
}